// CannyFilter_BPDA_2654289789548
// MI455X (gfx1250) — compile-verified
//
#include <hip/hip_runtime.h>
#include <hip/hip_bf16.h>
#include <math.h>
#include <stdint.h>

// Canny (BPDA) fully fused, one kernel per 32x32 output tile.
// img [32,3,512,512] f32 -> out [32,1,512,512] f32.
// gfx1250 async global->LDS loads (ASYNCcnt) stage the haloed tile;
// interior tiles use b128 (16B/lane) async loads, border tiles use clamped b32.

#define HW     512
#define CH     3
#define TS     32          // output tile edge
#define IMG_T  40          // TS + 2*4 halo (blur+sobel+nms+hyst)
#define BLR_T  38          // TS + 2*3
#define MAG_T  36          // TS + 2*2
#define THN_T  34          // TS + 2*1
#define NTHREADS 256

__device__ __forceinline__ int clampi(int v, int lo, int hi) {
    return v < lo ? lo : (v > hi ? hi : v);
}

__global__ __launch_bounds__(NTHREADS)
void canny_fused_kernel(const float* __restrict__ img,
                        const float* __restrict__ wg,     // 3x3 gaussian
                        const float* __restrict__ wsx,    // 3x3 sobel x
                        const float* __restrict__ wsy,    // 3x3 sobel y
                        const float* __restrict__ whyst,  // 3x3 (all 1.25)
                        float* __restrict__ out)
{
    __shared__ float         simg[CH][IMG_T][IMG_T]; // raw img tile, clamped coords
    __shared__ float         sblur[BLR_T][BLR_T];    // channel-summed blurred
    __shared__ float         smag[MAG_T][MAG_T];     // |grad| (0 outside image)
    __shared__ unsigned char sidx[MAG_T][MAG_T];     // orientation idx 0..7, 255=NaN
    __shared__ float         sthin[THN_T][THN_T];    // thin in {0,0.5,1} (0 outside)

    const int tid = threadIdx.x;
    const int X0  = blockIdx.x * TS;
    const int Y0  = blockIdx.y * TS;
    const int b   = blockIdx.z;

    // uniform 3x3 weights (scalar loads)
    float G[9], KX[9], KY[9];
#pragma unroll
    for (int k = 0; k < 9; ++k) { G[k] = wg[k]; KX[k] = wsx[k]; KY[k] = wsy[k]; }
    const float wh = whyst[4];

    // ---------------- Stage 0: async copy haloed img tile to LDS ----------------
    {
        const float*   base    = img + (size_t)b * CH * HW * HW; // uniform -> SGPR pair
        const unsigned ldsBase = (unsigned)(size_t)&simg[0][0][0];
        const bool interior = (X0 >= 4) && (X0 + TS + 4 <= HW) &&
                              (Y0 >= 4) && (Y0 + TS + 4 <= HW);
        if (interior) {
            // no clamping anywhere in the halo: rows contiguous; every 16B chunk
            // is 16B-aligned in both global ((X0-4)*4 is 16B-aligned) and LDS
            // (rows are 160B apart). 1200 x b128 chunks.
            const int NCHUNK = CH * IMG_T * (IMG_T / 4);       // 3*40*10 = 1200
            for (int v = tid; v < NCHUNK; v += NTHREADS) {
                int c  = v / (IMG_T * (IMG_T / 4));
                int r  = v - c * (IMG_T * (IMG_T / 4));
                int yy = r / (IMG_T / 4);
                int xq = r - yy * (IMG_T / 4);
                unsigned gOff = (unsigned)((((c * HW) + (Y0 - 4 + yy)) * HW
                                            + (X0 - 4) + xq * 4) * 4);
                unsigned lOff = ldsBase + (unsigned)(v * 16);
                asm volatile("global_load_async_to_lds_b128 %0, %1, %2"
                             :: "v"(lOff), "v"(gOff), "s"(base) : "memory");
            }
        } else {
            // replication clamp folded into the global address, 4B granules
            for (int e = tid; e < CH * IMG_T * IMG_T; e += NTHREADS) {
                int c  = e / (IMG_T * IMG_T);
                int r  = e - c * (IMG_T * IMG_T);
                int yy = r / IMG_T;
                int xx = r - yy * IMG_T;
                int gy = clampi(Y0 - 4 + yy, 0, HW - 1);
                int gx = clampi(X0 - 4 + xx, 0, HW - 1);
                unsigned gOff = (unsigned)((((c * HW) + gy) * HW + gx) * 4);
                unsigned lOff = ldsBase + (unsigned)(e * 4);
                asm volatile("global_load_async_to_lds_b32 %0, %1, %2"
                             :: "v"(lOff), "v"(gOff), "s"(base) : "memory");
            }
        }
        asm volatile("s_wait_asynccnt 0" ::: "memory");
    }
    __syncthreads();

    // ---------------- Stage 1: Gaussian blur, channel-summed --------------------
    // blur kernel identical per channel and Sobel sums channels, so sum channels
    // first: sblur = sum_c blur(img_c), evaluated at replication-clamped centers.
    for (int e = tid; e < BLR_T * BLR_T; e += NTHREADS) {
        int sy = e / BLR_T, sx = e - sy * BLR_T;
        int cy = clampi(Y0 - 3 + sy, 0, HW - 1);
        int cx = clampi(X0 - 3 + sx, 0, HW - 1);
        float acc = 0.f;
#pragma unroll
        for (int dy = -1; dy <= 1; ++dy) {
#pragma unroll
            for (int dx = -1; dx <= 1; ++dx) {
                int ty = clampi(cy + dy, 0, HW - 1) - (Y0 - 4);
                int tx = clampi(cx + dx, 0, HW - 1) - (X0 - 4);
                float s = simg[0][ty][tx] + simg[1][ty][tx] + simg[2][ty][tx];
                acc = fmaf(G[(dy + 1) * 3 + (dx + 1)], s, acc);
            }
        }
        sblur[sy][sx] = acc;
    }
    __syncthreads();

    // ---------------- Stage 2: Sobel -> magnitude + quantized orientation -------
    for (int e = tid; e < MAG_T * MAG_T; e += NTHREADS) {
        int my = e / MAG_T, mx = e - my * MAG_T;
        int gy = Y0 - 2 + my, gx = X0 - 2 + mx;
        float magv = 0.f;
        unsigned char idx = 255;                      // sentinel: NaN orientation
        if (gy >= 0 && gy < HW && gx >= 0 && gx < HW) {
            float ax = 0.f, ay = 0.f;
#pragma unroll
            for (int dy = -1; dy <= 1; ++dy) {
#pragma unroll
                for (int dx = -1; dx <= 1; ++dx) {
                    int sy = clampi(gy + dy, 0, HW - 1) - (Y0 - 3);
                    int sx = clampi(gx + dx, 0, HW - 1) - (X0 - 3);
                    float v = sblur[sy][sx];
                    int k = (dy + 1) * 3 + (dx + 1);
                    ax = fmaf(KX[k], v, ax);
                    ay = fmaf(KY[k], v, ay);
                }
            }
            ax = ax / 3.0f;                           // /C, as in reference
            ay = ay / 3.0f;
            magv = sqrtf(ax * ax + ay * ay);
            // ori = atan(gy/gx)*(360/pi)+180, rounded to 45 deg; idx = (ori/45)%8
            float ori = atanf(ay / ax) * 114.59155902616465f + 180.0f;
            float t = rintf(ori / 45.0f);             // RNE == jnp.round
            if (t == t) idx = (unsigned char)(((int)t) & 7);
        }
        smag[my][mx] = magv;                          // 0 outside image (zero pad)
        sidx[my][mx] = idx;
    }
    __syncthreads();

    // ---------------- Stage 3: directional NMS + double threshold ---------------
    // directional kernels are +1 center, -1 neighbour -> min(d[i],d[i+4])>0 is
    // exactly m > mag[fwd] && m > mag[bwd] along the quantized direction.
    for (int e = tid; e < THN_T * THN_T; e += NTHREADS) {
        int ty = e / THN_T, tx = e - ty * THN_T;
        int gy = Y0 - 1 + ty, gx = X0 - 1 + tx;
        float tv = 0.f;
        if (gy >= 0 && gy < HW && gx >= 0 && gx < HW) {
            const int my = ty + 1, mx = tx + 1;
            float m    = smag[my][mx];
            float thin = m;
            unsigned idx = sidx[my][mx];
            if (idx < 8) {
                // pos_i = idx&3 ; neighbour offsets (dy,dx) per pos_i, CCW from 'right'
                const int offy[4] = { 0, -1, -1, -1 };
                const int offx[4] = { 1,  1,  0, -1 };
                int p = (int)(idx & 3u);
                float fwd = smag[my + offy[p]][mx + offx[p]];
                float bwd = smag[my - offy[p]][mx - offx[p]];
                bool is_max = ((m - fwd) > 0.f) && ((m - bwd) > 0.f);
                if (!is_max) thin = 0.f;
            }
            float low  = (thin > 0.1f) ? 0.5f : 0.f;
            float high = (thin > 0.3f) ? 0.5f : 0.f;
            tv = low + high;                          // in {0, 0.5, 1}
        }
        sthin[ty][tx] = tv;                           // 0 outside image (zero pad)
    }
    __syncthreads();

    // ---------------- Stage 4: hysteresis + final combine -----------------------
    for (int e = tid; e < TS * TS; e += NTHREADS) {
        int oy = e / TS, ox = e - oy * TS;
        int ty = oy + 1, tx = ox + 1;
        float center = sthin[ty][tx];
        float s = 0.f;
#pragma unroll
        for (int dy = -1; dy <= 1; ++dy) {
#pragma unroll
            for (int dx = -1; dx <= 1; ++dx) {
                s += sthin[ty + dy][tx + dx];
            }
        }
        float weak1 = (s * wh > 1.0f) ? 1.f : 0.f;    // conv(thin,1.25*ones) > 1
        float high  = (center == 1.0f) ? 1.f : 0.f;
        float weak  = (center == 0.5f) ? 1.f : 0.f;
        size_t o = ((size_t)b * HW + (size_t)(Y0 + oy)) * HW + (size_t)(X0 + ox);
        out[o] = high + weak1 * weak;
    }
}

extern "C" void kernel_launch(void* const* d_in, const int* in_sizes, int n_in,
                              void* d_out, int out_size, void* d_ws, size_t ws_size,
                              hipStream_t stream) {
    (void)in_sizes; (void)n_in; (void)d_ws; (void)ws_size; (void)out_size;
    const float* img  = (const float*)d_in[0]; // [32,3,512,512]
    const float* wg   = (const float*)d_in[1]; // [1,1,3,3]
    const float* wsx  = (const float*)d_in[2]; // [1,1,3,3]
    const float* wsy  = (const float*)d_in[3]; // [1,1,3,3]
    // d_in[4] = weight_directional [8,1,3,3]: algebraically specialized in-kernel
    const float* whst = (const float*)d_in[5]; // [1,1,3,3] all 1.25
    float* out = (float*)d_out;                // [32,1,512,512]

    dim3 grid(HW / TS, HW / TS, 32);           // 16 x 16 x 32 tiles
    dim3 block(NTHREADS);                      // 8 wave32 waves
    canny_fused_kernel<<<grid, block, 0, stream>>>(img, wg, wsx, wsy, whst, out);
}